// FastLeFF_893353198364
// MI455X (gfx1250) — compile-verified
//
#include <hip/hip_runtime.h>
#include <math.h>

#define BATCH 2
#define DIM   32
#define HID   128
#define HH    64
#define WW    64
#define HWsz  4096

typedef float v2f __attribute__((ext_vector_type(2)));
typedef float v8f __attribute__((ext_vector_type(8)));
typedef unsigned int tdm_v4u __attribute__((ext_vector_type(4)));
typedef int tdm_v8i __attribute__((ext_vector_type(8)));
typedef int tdm_v4i __attribute__((ext_vector_type(4)));

__device__ __forceinline__ v8f wmma_f32(v2f a, v2f b, v8f c) {
  // D = A(16x4) * B(4x16) + C(16x16), fp32 WMMA
  return __builtin_amdgcn_wmma_f32_16x16x4_f32(false, a, false, b, (short)0, c, false, false);
}

__device__ __forceinline__ float gelu_f(float v) {
  return 0.5f * v * (1.0f + erff(v * 0.70710678118654752f));
}

// Issue a TDM 2D tile load: HID x HID f32 contiguous block -> LDS.
// D# layout per CDNA5 ISA ch.8: group0 {count|flags, lds_addr, global_addr_lo,
// global_addr_hi|type=2}, group1 {mask/data_size, dims, tile dims, strides}.
// This toolchain exposes the 6-arg builtin (extra int32x8 group; zero-filled).
__device__ __forceinline__ void tdm_load_tile_128x128(unsigned int lds_off, const float* src) {
  unsigned long long ga = (unsigned long long)(size_t)src;
  tdm_v4u g0;
  g0.x = 1u;                                   // count=1, is_restore=0, gather=0
  g0.y = lds_off;                              // LDS byte address
  g0.z = (unsigned int)ga;                     // global_addr[31:0]
  g0.w = ((unsigned int)(ga >> 32) & 0x01FFFFFFu) | (2u << 30);  // addr[56:32], type=2
  tdm_v8i g1;
  g1[0] = 0x20000;            // workgroup_mask=0, data_size=2 (4B), no flags
  g1[1] = (int)(HID << 16);   // tensor_dim0[15:0] in bits[63:48]
  g1[2] = (int)(HID << 16);   // tensor_dim0[31:16]=0, tensor_dim1[15:0]=128
  g1[3] = (int)(HID << 16);   // tensor_dim1 hi=0, tile_dim0=128
  g1[4] = HID;                // tile_dim1=128, tile_dim2=0
  g1[5] = HID;                // tensor_dim0_stride low32 = 128
  g1[6] = 0;                  // stride hi + dim1_stride low
  g1[7] = 0;
  tdm_v4i gz4 = {0, 0, 0, 0};
  tdm_v8i gz8 = {0, 0, 0, 0, 0, 0, 0, 0};
  __builtin_amdgcn_tensor_load_to_lds(g0, g1, gz4, gz4, gz8, 0);
}

// ---------------- Weight repack: (O,C,T) -> (T,C,O) ----------------
__global__ void repack_kernel(const float* __restrict__ src, float* __restrict__ dst,
                              int O, int C, int T) {
  int i = blockIdx.x * blockDim.x + threadIdx.x;
  if (i >= O * C * T) return;
  int t = i % T;
  int c = (i / T) % C;
  int o = i / (T * C);
  dst[((size_t)t * C + c) * O + o] = src[i];
}

// ---------------- GEMM1: x(B,HW,32) @ w1(32,128) + b1, gelu -> xi NCHW ----------------
__global__ void gemm1_kernel(const float* __restrict__ x, const float* __restrict__ w1,
                             const float* __restrict__ b1, float* __restrict__ xi) {
  int bid = blockIdx.x;
  int n_t = bid & 7;
  int m_t = (bid >> 3) & 255;
  int b   = bid >> 11;
  int lane = threadIdx.x;
  int ml = lane & 15;
  int sel = (lane < 16) ? 0 : 2;
  int m0 = m_t * 16, n0 = n_t * 16;
  v8f acc = {};
  const float* xrow = x + ((size_t)b * HWsz + (size_t)(m0 + ml)) * DIM;
  for (int kc = 0; kc < DIM; kc += 4) {
    v2f a = *(const v2f*)(xrow + kc + sel);
    v2f bb;
    bb.x = w1[(kc + sel) * HID + n0 + ml];
    bb.y = w1[(kc + sel + 1) * HID + n0 + ml];
    acc = wmma_f32(a, bb, acc);
  }
  int och = n0 + ml;
  float bias = b1[och];
  int mh = (lane >= 16) ? 8 : 0;
  for (int r = 0; r < 8; ++r) {
    int pix = m0 + r + mh;
    xi[((size_t)b * HID + och) * HWsz + pix] = gelu_f(acc[r] + bias);
  }
}

// ---------------- Depthwise 3x3 conv + bias ----------------
__global__ void dwconv_kernel(const float* __restrict__ xi, const float* __restrict__ w,
                              const float* __restrict__ bias, float* __restrict__ out) {
  int tid = blockIdx.x * blockDim.x + threadIdx.x;
  if (tid >= BATCH * HID * HWsz) return;
  int pix = tid & (HWsz - 1);
  int c = (tid >> 12) & (HID - 1);
  int b = tid >> 19;
  int y = pix >> 6, x = pix & 63;
  const float* in = xi + ((size_t)b * HID + c) * HWsz;
  const float* wc = w + c * 9;
  float s = bias[c];
  for (int ky = 0; ky < 3; ++ky) {
    int yy = y + ky - 1;
    if (yy < 0 || yy >= HH) continue;
    for (int kx = 0; kx < 3; ++kx) {
      int xx = x + kx - 1;
      if (xx < 0 || xx >= WW) continue;
      s += wc[ky * 3 + kx] * in[yy * WW + xx];
    }
  }
  out[tid] = s;
}

// ---------------- Generic direct conv as WMMA GEMM, weights repacked (T,Cin,Cout) ----------------
__global__ void conv_gemm_kernel(const float* __restrict__ in, int in_bstride,
                                 const float* __restrict__ w, const float* __restrict__ bias,
                                 float* __restrict__ out, int out_bstride,
                                 int Cin, int Cout, int K, int pad, int Ntiles, int do_gelu) {
  int bid = blockIdx.x;
  int n_t = bid % Ntiles;
  int m_t = (bid / Ntiles) & 255;
  int b = bid / (Ntiles * 256);
  int lane = threadIdx.x;
  int ml = lane & 15;
  int sel = (lane < 16) ? 0 : 2;
  int m0 = m_t * 16, n0 = n_t * 16;
  int y = m0 >> 6, x0 = m0 & 63;
  int och = n0 + ml;
  int ochl = (och < Cout) ? och : (Cout - 1);
  v8f acc = {};
  for (int ky = 0; ky < K; ++ky) {
    int yy = y + ky - pad;
    bool rowok = (yy >= 0) && (yy < HH);
    int yyc = yy < 0 ? 0 : (yy >= HH ? HH - 1 : yy);
    for (int kx = 0; kx < K; ++kx) {
      int xx = x0 + ml + kx - pad;
      bool ok = rowok && (xx >= 0) && (xx < WW);
      int xxc = xx < 0 ? 0 : (xx >= WW ? WW - 1 : xx);
      const float* ip = in + (size_t)b * in_bstride + (size_t)yyc * WW + xxc;
      const float* wp = w + (size_t)(ky * K + kx) * Cin * Cout + ochl;   // [t][c][o]
      for (int kc = 0; kc < Cin; kc += 4) {
        int c = kc + sel;
        v2f a;
        a.x = ok ? ip[(size_t)c * HWsz] : 0.0f;
        a.y = ok ? ip[(size_t)(c + 1) * HWsz] : 0.0f;
        v2f bb;
        bb.x = wp[(size_t)c * Cout];
        bb.y = wp[(size_t)(c + 1) * Cout];
        acc = wmma_f32(a, bb, acc);
      }
    }
  }
  if (och < Cout) {
    float bv = bias[och];
    int mh = (lane >= 16) ? 8 : 0;
    for (int r = 0; r < 8; ++r) {
      int pix = m0 + r + mh;
      float v = acc[r] + bv;
      out[(size_t)b * out_bstride + (size_t)och * HWsz + pix] = do_gelu ? gelu_f(v) : v;
    }
  }
}

// ---------------- Fused deformable conv ----------------
// Block = 256 threads (8 waves). Per tap: TDM stages the 128x128 weight slice
// (repacked [t][c][o]) into LDS while threads compute the bilinear-sampled
// S tile (16px x 128ch) into LDS; then 8 waves run 16x16 WMMA tiles from LDS.
__global__ void deform_kernel(const float* __restrict__ ain, int a_bstride,
                              const float* __restrict__ off, int off_bstride,
                              const float* __restrict__ wt,   // [T][HID][HID]
                              const float* __restrict__ bias,
                              float* __restrict__ out, int out_bstride,
                              int K, int pad) {
  __shared__ float S[16][132];           // sampled tile
  __shared__ float Wbuf[HID * HID];      // per-tap weight slice (64KB) via TDM
  int bid = blockIdx.x;
  int m_t = bid & 255;
  int b = bid >> 8;
  int m0 = m_t * 16;
  int y = m0 >> 6, x0 = m0 & 63;
  int T = K * K;
  int tid = threadIdx.x;
  int lane = tid & 31;
  int wv = tid >> 5;
  int ml = lane & 15;
  int sel = (lane < 16) ? 0 : 2;
  int n0 = wv * 16;
  int p = tid & 15;
  int c0 = (tid >> 4) * 8;
  int px = x0 + p;
  const float* abase = ain + (size_t)b * a_bstride;
  unsigned int wbuf_off = (unsigned int)(size_t)(void*)Wbuf;  // LDS byte offset (low 32 bits)
  v8f acc = {};
  for (int t = 0; t < T; ++t) {
    __syncthreads();                       // prior tap's LDS reads complete
    if (wv == 0) {
      tdm_load_tile_128x128(wbuf_off, wt + (size_t)t * HID * HID);
    }
    int ky = t / K, kx = t - ky * K;
    float dy = off[(size_t)b * off_bstride + (size_t)(2 * t) * HWsz + y * WW + px];
    float dx = off[(size_t)b * off_bstride + (size_t)(2 * t + 1) * HWsz + y * WW + px];
    float py = (float)(y + ky - pad) + dy;
    float pxf = (float)(px + kx - pad) + dx;
    float y0f = floorf(py), x0f = floorf(pxf);
    float wy = py - y0f, wx = pxf - x0f;
    int iy0 = (int)y0f, ix0 = (int)x0f;
    int iy1 = iy0 + 1, ix1 = ix0 + 1;
    bool vy0 = (iy0 >= 0) && (iy0 < HH);
    bool vy1 = (iy1 >= 0) && (iy1 < HH);
    bool vx0 = (ix0 >= 0) && (ix0 < WW);
    bool vx1 = (ix1 >= 0) && (ix1 < WW);
    float w00 = (vy0 && vx0) ? (1.f - wy) * (1.f - wx) : 0.f;
    float w01 = (vy0 && vx1) ? (1.f - wy) * wx : 0.f;
    float w10 = (vy1 && vx0) ? wy * (1.f - wx) : 0.f;
    float w11 = (vy1 && vx1) ? wy * wx : 0.f;
    int cy0 = iy0 < 0 ? 0 : (iy0 >= HH ? HH - 1 : iy0);
    int cy1 = iy1 < 0 ? 0 : (iy1 >= HH ? HH - 1 : iy1);
    int cx0 = ix0 < 0 ? 0 : (ix0 >= WW ? WW - 1 : ix0);
    int cx1 = ix1 < 0 ? 0 : (ix1 >= WW ? WW - 1 : ix1);
    int i00 = cy0 * WW + cx0, i01 = cy0 * WW + cx1;
    int i10 = cy1 * WW + cx0, i11 = cy1 * WW + cx1;
    for (int i = 0; i < 8; ++i) {
      const float* cp = abase + (size_t)(c0 + i) * HWsz;
      S[p][c0 + i] = w00 * cp[i00] + w01 * cp[i01] + w10 * cp[i10] + w11 * cp[i11];
    }
    if (wv == 0) {
      __builtin_amdgcn_s_wait_tensorcnt(0);   // DMA complete in issuing wave
    }
    __syncthreads();                          // publish S + Wbuf to all waves
    for (int kc = 0; kc < HID; kc += 4) {
      v2f a = *(const v2f*)&S[ml][kc + sel];           // ds_load_b64
      v2f bb;
      bb.x = Wbuf[(kc + sel) * HID + n0 + ml];          // ds_load_b32
      bb.y = Wbuf[(kc + sel + 1) * HID + n0 + ml];
      acc = wmma_f32(a, bb, acc);
    }
  }
  int och = n0 + ml;
  float bv = bias[och];
  int mh = (lane >= 16) ? 8 : 0;
  for (int r = 0; r < 8; ++r) {
    out[(size_t)b * out_bstride + (size_t)och * HWsz + (m0 + r + mh)] = gelu_f(acc[r] + bv);
  }
}

// ---------------- Final: z = gelu(a)*g ; z @ w2(128,32) + b2 -> (B,HW,32) ----------------
__global__ void final_kernel(const float* __restrict__ a3, const float* __restrict__ yg,
                             const float* __restrict__ w2, const float* __restrict__ b2,
                             float* __restrict__ out) {
  int bid = blockIdx.x;
  int n_t = bid & 1;
  int m_t = (bid >> 1) & 255;
  int b = bid >> 9;
  int lane = threadIdx.x;
  int ml = lane & 15;
  int sel = (lane < 16) ? 0 : 2;
  int m0 = m_t * 16, n0 = n_t * 16;
  int pix = m0 + ml;
  const float* ap = a3 + (size_t)b * HID * HWsz + pix;
  const float* gp = yg + (size_t)b * 2 * HID * HWsz + (size_t)HID * HWsz + pix;
  v8f acc = {};
  for (int kc = 0; kc < HID; kc += 4) {
    int c = kc + sel;
    v2f a;
    a.x = gelu_f(ap[(size_t)c * HWsz]) * gp[(size_t)c * HWsz];
    a.y = gelu_f(ap[(size_t)(c + 1) * HWsz]) * gp[(size_t)(c + 1) * HWsz];
    v2f bb;
    bb.x = w2[c * DIM + n0 + ml];
    bb.y = w2[(c + 1) * DIM + n0 + ml];
    acc = wmma_f32(a, bb, acc);
  }
  int och = n0 + ml;
  float bv = b2[och];
  int mh = (lane >= 16) ? 8 : 0;
  for (int r = 0; r < 8; ++r) {
    int pp = m0 + r + mh;
    out[((size_t)b * HWsz + pp) * DIM + och] = acc[r] + bv;
  }
}

static inline int cdiv(int a, int b) { return (a + b - 1) / b; }

extern "C" void kernel_launch(void* const* d_in, const int* in_sizes, int n_in,
                              void* d_out, int out_size, void* d_ws, size_t ws_size,
                              hipStream_t stream) {
  const float* x    = (const float*)d_in[0];
  const float* w1   = (const float*)d_in[1];
  const float* b1   = (const float*)d_in[2];
  const float* dw_w = (const float*)d_in[3];
  const float* dw_b = (const float*)d_in[4];
  const float* pw_w = (const float*)d_in[5];
  const float* pw_b = (const float*)d_in[6];
  const float* d3ow = (const float*)d_in[7];
  const float* d3ob = (const float*)d_in[8];
  const float* d3w  = (const float*)d_in[9];
  const float* d3b  = (const float*)d_in[10];
  const float* d4ow = (const float*)d_in[11];
  const float* d4ob = (const float*)d_in[12];
  const float* d4w  = (const float*)d_in[13];
  const float* d4b  = (const float*)d_in[14];
  const float* d5ow = (const float*)d_in[15];
  const float* d5ob = (const float*)d_in[16];
  const float* d5w  = (const float*)d_in[17];
  const float* d5b  = (const float*)d_in[18];
  const float* w2   = (const float*)d_in[19];
  const float* b2   = (const float*)d_in[20];
  float* out = (float*)d_out;

  const size_t N1 = (size_t)BATCH * HID * HWsz;
  const size_t N2 = (size_t)BATCH * 2 * HID * HWsz;
  const size_t NOFF = (size_t)BATCH * 98 * HWsz;
  float* ws   = (float*)d_ws;
  float* xi   = ws;
  float* yd   = xi + N1;
  float* y    = yd + N1;
  float* offb = y + N2;
  float* a1   = offb + NOFF;
  float* a2   = a1 + N1;
  float* a3   = a2 + N1;
  float* wr   = a3 + N1;   // shared repack buffer (stream-ordered reuse)

  gemm1_kernel<<<dim3(BATCH * 256 * 8), dim3(32), 0, stream>>>(x, w1, b1, xi);
  dwconv_kernel<<<dim3((BATCH * HID * HWsz) / 256), dim3(256), 0, stream>>>(xi, dw_w, dw_b, yd);

  // pointwise 1x1 -> 256 ch, gelu
  repack_kernel<<<dim3(cdiv(256 * 128 * 1, 256)), dim3(256), 0, stream>>>(pw_w, wr, 256, 128, 1);
  conv_gemm_kernel<<<dim3(BATCH * 256 * 16), dim3(32), 0, stream>>>(
      yd, HID * HWsz, wr, pw_b, y, 2 * HID * HWsz, HID, 2 * HID, 1, 0, 16, 1);

  // ---- d3 (k=7) ----
  repack_kernel<<<dim3(cdiv(98 * 128 * 49, 256)), dim3(256), 0, stream>>>(d3ow, wr, 98, 128, 49);
  conv_gemm_kernel<<<dim3(BATCH * 256 * 7), dim3(32), 0, stream>>>(
      y, 2 * HID * HWsz, wr, d3ob, offb, 98 * HWsz, HID, 98, 7, 3, 7, 0);
  repack_kernel<<<dim3(cdiv(128 * 128 * 49, 256)), dim3(256), 0, stream>>>(d3w, wr, 128, 128, 49);
  deform_kernel<<<dim3(BATCH * 256), dim3(256), 0, stream>>>(
      y, 2 * HID * HWsz, offb, 98 * HWsz, wr, d3b, a1, HID * HWsz, 7, 3);

  // ---- d4 (k=3) ----
  repack_kernel<<<dim3(cdiv(18 * 128 * 9, 256)), dim3(256), 0, stream>>>(d4ow, wr, 18, 128, 9);
  conv_gemm_kernel<<<dim3(BATCH * 256 * 2), dim3(32), 0, stream>>>(
      a1, HID * HWsz, wr, d4ob, offb, 18 * HWsz, HID, 18, 3, 1, 2, 0);
  repack_kernel<<<dim3(cdiv(128 * 128 * 9, 256)), dim3(256), 0, stream>>>(d4w, wr, 128, 128, 9);
  deform_kernel<<<dim3(BATCH * 256), dim3(256), 0, stream>>>(
      a1, HID * HWsz, offb, 18 * HWsz, wr, d4b, a2, HID * HWsz, 3, 1);

  // ---- d5 (k=5) ----
  repack_kernel<<<dim3(cdiv(50 * 128 * 25, 256)), dim3(256), 0, stream>>>(d5ow, wr, 50, 128, 25);
  conv_gemm_kernel<<<dim3(BATCH * 256 * 4), dim3(32), 0, stream>>>(
      a2, HID * HWsz, wr, d5ob, offb, 50 * HWsz, HID, 50, 5, 2, 4, 0);
  repack_kernel<<<dim3(cdiv(128 * 128 * 25, 256)), dim3(256), 0, stream>>>(d5w, wr, 128, 128, 25);
  deform_kernel<<<dim3(BATCH * 256), dim3(256), 0, stream>>>(
      a2, HID * HWsz, offb, 50 * HWsz, wr, d5b, a3, HID * HWsz, 5, 2);

  // ---- gate + GEMM2 ----
  final_kernel<<<dim3(BATCH * 256 * 2), dim3(32), 0, stream>>>(a3, y, w2, b2, out);

  (void)in_sizes; (void)n_in; (void)out_size; (void)ws_size;
}